// Net_26671746908314
// MI455X (gfx1250) — compile-verified
//
#include <hip/hip_runtime.h>
#include <hip/hip_bf16.h>
#include <math.h>

typedef float v2f __attribute__((ext_vector_type(2)));
typedef float v8f __attribute__((ext_vector_type(8)));

#define EPSBN 1e-6f

// ---------------------------------------------------------------------------
// Tiled GEMM with WMMA f32 16x16x4.
// X[M x Kd] @ W[Kd x Nout] (+ bias) -> Y[M x Nout].  REQUIRES M % 16 == 0.
// Workgroup = 4 waves = 4 consecutive M-tiles sharing one N-tile.
// W tile is staged to LDS once (clamped + zero-padded), so the inner loop is
// a merged global_load_b64 (A) + two ds_loads (B) + WMMA, double-buffered.
// Optionally accumulates per-channel sum/sumsq into stats[0..127]/[128..255].
// ---------------------------------------------------------------------------
#define GEMM_KMAX 128
__global__ __launch_bounds__(128) void gemm_wmma_kernel(
    const float* __restrict__ X, int M, int Kd,
    const float* __restrict__ W, const float* __restrict__ bias,
    float* __restrict__ Y, int Nout,
    float* __restrict__ stats)
{
    __shared__ float sW[GEMM_KMAX * 16];     // [k][n] tile, Kpad <= 128
    const int tile_n = blockIdx.y * 16;
    const int tid = threadIdx.x;
    const int Kpad = (Kd + 3) & ~3;

    // cooperative stage of W tile: clamped loads, zero-padded cols and K rows
    for (int t = tid; t < Kpad * 16; t += 128) {
        int k = t >> 4;
        int n = tile_n + (t & 15);
        bool ok = (n < Nout) && (k < Kd);
        int kc = (k < Kd) ? k : (Kd - 1);
        int nc = (n < Nout) ? n : (Nout - 1);
        float wv = W[(size_t)kc * Nout + nc];
        sW[t] = ok ? wv : 0.0f;
    }
    __syncthreads();

    const int wv_  = tid >> 5;
    const int lane = tid & 31;
    const int hl   = lane >> 4;
    const int l16  = lane & 15;
    const int tile_m = (blockIdx.x * 4 + wv_) * 16;
    if (tile_m >= M) return;                 // wave-uniform exit (EXEC stays full)

    const int m_a = tile_m + l16;            // always < M (M % 16 == 0)
    const int n_b = tile_n + l16;
    const bool ncol = (n_b < Nout);

    const float* __restrict__ Xr = X + (size_t)m_a * Kd;
    const int kbody = Kd & ~3;

    v8f c = {};
    if (kbody > 0) {
        v2f a0, b0;
        {
            int k0 = 2 * hl;
            a0[0] = Xr[k0];
            a0[1] = Xr[k0 + 1];
            b0[0] = sW[k0 * 16 + l16];
            b0[1] = sW[(k0 + 1) * 16 + l16];
        }
        int sws = 2 * hl * 16 + l16;         // LDS offset walk: +64 per K-step
        for (int kk = 4; kk < kbody; kk += 4) {
            v2f a1, b1;
            int k1 = kk + 2 * hl;
            a1[0] = Xr[k1];
            a1[1] = Xr[k1 + 1];
            b1[0] = sW[sws + 64];
            b1[1] = sW[sws + 80];
            sws += 64;
            c = __builtin_amdgcn_wmma_f32_16x16x4_f32(false, a0, false, b0,
                                                      (short)0, c, false, false);
            a0 = a1; b0 = b1;
        }
        c = __builtin_amdgcn_wmma_f32_16x16x4_f32(false, a0, false, b0,
                                                  (short)0, c, false, false);
    }
    if (kbody < Kd) {               // K tail (only fc0's Kd=6 reaches this)
        v2f a, b;
#pragma unroll
        for (int v = 0; v < 2; ++v) {
            int k = kbody + v + 2 * hl;
            int kc = (k < Kd) ? k : (Kd - 1);
            float xa = Xr[kc];
            a[v] = (k < Kd) ? xa : 0.0f;
            b[v] = sW[k * 16 + l16];         // pad rows are already zero
        }
        c = __builtin_amdgcn_wmma_f32_16x16x4_f32(false, a, false, b,
                                                  (short)0, c, false, false);
    }

    if (ncol) {
        float bv = (bias != nullptr) ? bias[n_b] : 0.0f;
        float psum = 0.0f, psq = 0.0f;
#pragma unroll
        for (int r = 0; r < 8; ++r) {
            int m = tile_m + r + 8 * hl;            // ISA 16x16 C/D layout
            float y = c[r] + bv;
            Y[(size_t)m * Nout + n_b] = y;
            psum += y;
            psq  += y * y;
        }
        if (stats != nullptr) {
            atomicAdd(&stats[n_b], psum);
            atomicAdd(&stats[n_b + 128], psq);
        }
    }
}

// ---------------------------------------------------------------------------
// Zero a small float region (BN stats scratch).
// ---------------------------------------------------------------------------
__global__ void zero_kernel(float* __restrict__ p, int n)
{
    int i = blockIdx.x * blockDim.x + threadIdx.x;
    if (i < n) p[i] = 0.0f;
}

// ---------------------------------------------------------------------------
// Finalize BN: stats[c]=sum, stats[c+128]=sumsq  ->  scale at +256, shift +384.
// ---------------------------------------------------------------------------
__global__ void finalize_bn_kernel(float* __restrict__ stats,
                                   const float* __restrict__ g,
                                   const float* __restrict__ beta,
                                   int C, float invCnt)
{
    int c = threadIdx.x;
    if (c < C) {
        float mean = stats[c] * invCnt;
        float var  = stats[c + 128] * invCnt - mean * mean;
        float rstd = rsqrtf(var + EPSBN);
        float sc   = g[c] * rstd;
        stats[c + 256] = sc;
        stats[c + 384] = beta[c] - mean * sc;
    }
}

// ---------------------------------------------------------------------------
// BN apply + activation / residual. mode: 0 = none, 1 = lrelu,
// 2 = add residual then lrelu. cmask = C-1 (C is a power of two here).
// ---------------------------------------------------------------------------
__global__ void bn_act_kernel(float* __restrict__ Y,
                              const float* __restrict__ stats,
                              int total, int cmask, int mode,
                              const float* __restrict__ resid)
{
    int gid = blockIdx.x * blockDim.x + threadIdx.x;
    if (gid >= total) return;
    int c = gid & cmask;
    float y = Y[gid] * stats[c + 256] + stats[c + 384];
    if (mode >= 1) {
        if (mode == 2) y += resid[gid];
        y = (y >= 0.0f) ? y : 0.2f * y;
    }
    Y[gid] = y;
}

// ---------------------------------------------------------------------------
// KNN: per query point, branchless register top-16 (stable, matches top_k of
// -d2 with ties resolved by index order), candidates staged through LDS,
// next tile prefetched with global_prefetch.
// ---------------------------------------------------------------------------
#define KNN_TILE 128
__global__ __launch_bounds__(128) void knn_kernel(
    const float* __restrict__ pos, int Bn, int* __restrict__ idx)
{
    __shared__ float sp[KNN_TILE * 3];
    int b = blockIdx.y;
    int i = blockIdx.x * blockDim.x + threadIdx.x;
    const float* pb = pos + (size_t)b * Bn * 3;
    bool valid = (i < Bn);
    float px = 0.f, py = 0.f, pz = 0.f;
    if (valid) { px = pb[i * 3 + 0]; py = pb[i * 3 + 1]; pz = pb[i * 3 + 2]; }

    float bestd[16];
    int   besti[16];
#pragma unroll
    for (int k = 0; k < 16; ++k) { bestd[k] = 3.4e38f; besti[k] = 0; }

    for (int j0 = 0; j0 < Bn; j0 += KNN_TILE) {
        int jl = j0 + threadIdx.x;
        if (jl < Bn) {
            sp[threadIdx.x * 3 + 0] = pb[jl * 3 + 0];
            sp[threadIdx.x * 3 + 1] = pb[jl * 3 + 1];
            sp[threadIdx.x * 3 + 2] = pb[jl * 3 + 2];
        }
        if (j0 + KNN_TILE < Bn)
            __builtin_prefetch(&pb[(j0 + KNN_TILE + (int)threadIdx.x) * 3], 0, 1);
        __syncthreads();
        int jend = (Bn - j0 < KNN_TILE) ? (Bn - j0) : KNN_TILE;
        if (valid) {
            for (int j = 0; j < jend; ++j) {
                float dx = px - sp[j * 3 + 0];
                float dy = py - sp[j * 3 + 1];
                float dz = pz - sp[j * 3 + 2];
                float d = dx * dx + dy * dy + dz * dz;
                if (d < bestd[15]) {
                    float cd = d; int ci = j0 + j;
#pragma unroll
                    for (int k = 0; k < 16; ++k) {
                        bool lt = cd < bestd[k];
                        float td = bestd[k]; int ti = besti[k];
                        bestd[k] = lt ? cd : td;
                        besti[k] = lt ? ci : ti;
                        cd = lt ? td : cd;
                        ci = lt ? ti : ci;
                    }
                }
            }
        }
        __syncthreads();
    }
    if (valid) {
#pragma unroll
        for (int k = 0; k < 16; ++k)
            idx[((size_t)b * Bn + i) * 16 + k] = besti[k];
    }
}

// ---------------------------------------------------------------------------
// LFA phase 1: accumulate BN stats of enc pre-activation over (B,N,K).
// One thread per (point, neighbor). LDS partial sums -> global atomics.
// ---------------------------------------------------------------------------
__global__ __launch_bounds__(256) void lfa_enc_stats_kernel(
    const float* __restrict__ pos, const int* __restrict__ idx,
    int bnShift, int Mtot,
    const float* __restrict__ encW, const float* __restrict__ encB,
    int d2, float* __restrict__ stats)
{
    __shared__ float ssum[32], ssq[32];
    if (threadIdx.x < 32) { ssum[threadIdx.x] = 0.0f; ssq[threadIdx.x] = 0.0f; }
    __syncthreads();

    int gid = blockIdx.x * blockDim.x + threadIdx.x;
    if (gid < Mtot * 16) {
        int p = gid >> 4;
        int k = gid & 15;
        int b = p >> bnShift;
        int j = idx[p * 16 + k];
        int rowj = (b << bnShift) + j;
        float pix = pos[p * 3 + 0], piy = pos[p * 3 + 1], piz = pos[p * 3 + 2];
        float pjx = pos[rowj * 3 + 0], pjy = pos[rowj * 3 + 1], pjz = pos[rowj * 3 + 2];
        float dx = pjx - pix, dy = pjy - piy, dz = pjz - piz;
        float dist = sqrtf(dx * dx + dy * dy + dz * dz);
        float rel[10] = { pix, piy, piz, pjx, pjy, pjz, dx, dy, dz, dist };
        for (int c = 0; c < d2; ++c) {
            float acc = encB[c];
#pragma unroll
            for (int t = 0; t < 10; ++t) acc += rel[t] * encW[t * d2 + c];
            atomicAdd(&ssum[c], acc);
            atomicAdd(&ssq[c], acc * acc);
        }
    }
    __syncthreads();
    if (threadIdx.x < (unsigned)d2) {
        atomicAdd(&stats[threadIdx.x], ssum[threadIdx.x]);
        atomicAdd(&stats[threadIdx.x + 128], ssq[threadIdx.x]);
    }
}

// ---------------------------------------------------------------------------
// LFA fused main pass. One wave = one point.
//   att_W staged to LDS with async global->LDS B128 copies (ASYNCcnt)
//   feats[K=16][d] = [gathered x_j | bn+lrelu(enc(rel))] in LDS
//   scores = feats @ att_W via WMMA f32 16x16x4
//   softmax over K per channel, agg = sum_k att*feats
//   post_pre = agg @ postW + postB; BN stats reduced in LDS, one global
//   atomic per channel per workgroup.
// ---------------------------------------------------------------------------
#define LFA_MAXD 64
__global__ __launch_bounds__(128) void lfa_main_kernel(
    const float* __restrict__ x,        // Mtot x d2
    const float* __restrict__ pos,      // Mtot x 3
    const int*   __restrict__ idx,      // Mtot x 16
    int bnShift, int Mtot, int d,
    const float* __restrict__ encW,     // 10 x d2
    const float* __restrict__ encB,     // d2
    const float* __restrict__ encStats, // scale at +256, shift at +384
    const float* __restrict__ attW,     // d x d
    const float* __restrict__ postW,    // d x d
    const float* __restrict__ postB,    // d
    float* __restrict__ outPre,         // Mtot x d
    float* __restrict__ postStats)
{
    __shared__ float sAtt[LFA_MAXD * LFA_MAXD];
    __shared__ float sFeat[4][16 * LFA_MAXD];
    __shared__ float sScore[4][16 * LFA_MAXD];
    __shared__ float sAgg[4][LFA_MAXD];
    __shared__ float sSum[LFA_MAXD], sSq[LFA_MAXD];

    int d2 = d >> 1;

    // async global->LDS staging of att_W (d*d is a multiple of 16 floats)
    for (int t = threadIdx.x * 4; t < d * d; t += 128 * 4) {
        unsigned lds_off = (unsigned)(uintptr_t)(&sAtt[t]);
        const float* gp = attW + t;
        asm volatile("global_load_async_to_lds_b128 %0, %1, off"
                     :: "v"(lds_off), "v"(gp) : "memory");
    }
    if (threadIdx.x < LFA_MAXD) { sSum[threadIdx.x] = 0.0f; sSq[threadIdx.x] = 0.0f; }

    int w    = threadIdx.x >> 5;
    int lane = threadIdx.x & 31;
    int hl   = lane >> 4;
    int l16  = lane & 15;
    int p = blockIdx.x * 4 + w;
    bool pvalid = (p < Mtot);
    int pp = pvalid ? p : (Mtot - 1);
    int b = pp >> bnShift;

    // ---- build feats in LDS --------------------------------------------
    {
        int k = l16;
        int j = idx[pp * 16 + k];
        int rowj = (b << bnShift) + j;
        if (hl == 0) {
            for (int c = 0; c < d2; ++c)
                sFeat[w][k * d + c] = x[(size_t)rowj * d2 + c];
        } else {
            float pix = pos[pp * 3 + 0], piy = pos[pp * 3 + 1], piz = pos[pp * 3 + 2];
            float pjx = pos[rowj * 3 + 0], pjy = pos[rowj * 3 + 1], pjz = pos[rowj * 3 + 2];
            float dx = pjx - pix, dy = pjy - piy, dz = pjz - piz;
            float dist = sqrtf(dx * dx + dy * dy + dz * dz);
            float rel[10] = { pix, piy, piz, pjx, pjy, pjz, dx, dy, dz, dist };
            for (int c = 0; c < d2; ++c) {
                float acc = encB[c];
#pragma unroll
                for (int t = 0; t < 10; ++t) acc += rel[t] * encW[t * d2 + c];
                float y = acc * encStats[256 + c] + encStats[384 + c];
                y = (y >= 0.0f) ? y : 0.2f * y;
                sFeat[w][k * d + d2 + c] = y;
            }
        }
    }
    asm volatile("s_wait_asynccnt 0x0" ::: "memory");
    __syncthreads();

    // ---- scores = feats @ attW via WMMA --------------------------------
    for (int n0 = 0; n0 < d; n0 += 16) {
        v8f cacc = {};
        int n = n0 + l16;
        int n_c = (n < d) ? n : (d - 1);
        bool ncol = (n < d);
        for (int kk = 0; kk < d; kk += 4) {
            v2f av, bv;
#pragma unroll
            for (int v = 0; v < 2; ++v) {
                int kc = kk + v + 2 * hl;
                av[v] = sFeat[w][l16 * d + kc];
                float wv = sAtt[kc * d + n_c];
                bv[v] = ncol ? wv : 0.0f;
            }
            cacc = __builtin_amdgcn_wmma_f32_16x16x4_f32(false, av, false, bv,
                                                         (short)0, cacc, false, false);
        }
        if (ncol) {
#pragma unroll
            for (int r = 0; r < 8; ++r)
                sScore[w][(r + 8 * hl) * d + n] = cacc[r];
        }
    }
    __syncthreads();

    // ---- softmax over K (axis=2) per channel, aggregate -----------------
    for (int c = lane; c < d; c += 32) {
        float mx = -3.4e38f;
#pragma unroll
        for (int k = 0; k < 16; ++k) mx = fmaxf(mx, sScore[w][k * d + c]);
        float e[16];
        float s = 0.0f;
#pragma unroll
        for (int k = 0; k < 16; ++k) {
            e[k] = __expf(sScore[w][k * d + c] - mx);
            s += e[k];
        }
        float inv = 1.0f / s;
        float acc = 0.0f;
#pragma unroll
        for (int k = 0; k < 16; ++k) acc += e[k] * inv * sFeat[w][k * d + c];
        sAgg[w][c] = acc;
    }
    __syncthreads();

    // ---- post linear (pre-BN) + LDS-reduced stats -----------------------
    for (int co = lane; co < d; co += 32) {
        float acc = postB[co];
        for (int ci = 0; ci < d; ++ci) acc += sAgg[w][ci] * postW[ci * d + co];
        if (pvalid) {
            outPre[(size_t)pp * d + co] = acc;
            atomicAdd(&sSum[co], acc);
            atomicAdd(&sSq[co], acc * acc);
        }
    }
    __syncthreads();
    if (threadIdx.x < (unsigned)d) {
        atomicAdd(&postStats[threadIdx.x], sSum[threadIdx.x]);
        atomicAdd(&postStats[threadIdx.x + 128], sSq[threadIdx.x]);
    }
}

// ---------------------------------------------------------------------------
// Row slicing for the decimation steps: dst[b, 0:dstN, :] = src[b, 0:dstN, :]
// ---------------------------------------------------------------------------
__global__ void slice_rows_kernel(const float* __restrict__ src, int srcN,
                                  float* __restrict__ dst, int dstN, int C)
{
    int gid = blockIdx.x * blockDim.x + threadIdx.x;
    int total = 16 * dstN * C;
    if (gid >= total) return;
    int c = gid % C;
    int r = (gid / C) % dstN;
    int b = gid / (C * dstN);
    dst[((size_t)b * dstN + r) * C + c] = src[((size_t)b * srcN + r) * C + c];
}

// ---------------------------------------------------------------------------
// Global max pool over points: g[b,c] = max_r h[b,r,c]
// ---------------------------------------------------------------------------
__global__ __launch_bounds__(128) void maxpool_kernel(
    const float* __restrict__ h, int Np, int C, float* __restrict__ g)
{
    int b = blockIdx.x;
    int c = threadIdx.x;
    if (c >= C) return;
    float m = -3.4e38f;
    for (int r = 0; r < Np; ++r)
        m = fmaxf(m, h[((size_t)b * Np + r) * C + c]);
    g[b * C + c] = m;
}

// ---------------------------------------------------------------------------
// Row-wise log_softmax over 40 logits, 16 rows.
// ---------------------------------------------------------------------------
__global__ void logsoftmax_kernel(const float* __restrict__ z, float* __restrict__ out)
{
    int b = threadIdx.x;
    if (b >= 16) return;
    float mx = -3.4e38f;
    for (int c = 0; c < 40; ++c) mx = fmaxf(mx, z[b * 40 + c]);
    float s = 0.0f;
    for (int c = 0; c < 40; ++c) s += __expf(z[b * 40 + c] - mx);
    float lse = mx + logf(s);
    for (int c = 0; c < 40; ++c) out[b * 40 + c] = z[b * 40 + c] - lse;
}

// ---------------------------------------------------------------------------
// Host orchestration.
// Param index map (JAX pytree flatten: dict keys sorted, DFS):
//   0:x 1:pos
//   block1 @2:  lfa1{att=+0, enc W/b/beta/g=+1..+4, post W/b/beta/g=+5..+8},
//               lfa2 @+9, mlp1 W/b/beta/g @+18, mlp2 @+22, shortcut @+26
//   block2 @32 (same layout)
//   end0 @62 (W,b,beta,g)   fc0 @66 (W,b)   fc_end @68 (W,b)   mlp1 @70 (W,b,beta,g)
// ---------------------------------------------------------------------------
extern "C" void kernel_launch(void* const* d_in, const int* in_sizes, int n_in,
                              void* d_out, int out_size, void* d_ws, size_t ws_size,
                              hipStream_t stream)
{
    (void)in_sizes; (void)n_in; (void)out_size; (void)ws_size;
    auto P = [&](int i) -> const float* { return (const float*)d_in[i]; };
    const float* X0  = P(0);
    const float* POS = P(1);

    float* ws   = (float*)d_ws;
    int*   IDX  = (int*)ws;                 // 32768*16 ints
    float* HA   = ws + 524288;              // 1,048,576 f
    float* HB   = HA + 1048576;             // 1,048,576 f
    float* SC   = HB + 1048576;             // 1,048,576 f
    float* TMPA = SC + 1048576;             //   524,288 f
    float* TMPB = TMPA + 524288;            //   524,288 f
    float* POS2 = TMPB + 524288;            //    24,576 f
    float* STA  = POS2 + 24576;             //       512 f (enc BN stats)
    float* STB  = STA + 512;                //       512 f (smlp/post BN stats)
    float* G1   = STB + 512;                //      2048 f
    float* G2   = G1 + 2048;                //       512 f
    float* G3   = G2 + 512;                 //       640 f

    auto gemm = [&](const float* Xp, int M, int Kd, const float* W,
                    const float* bias, float* Y, int Nout, float* stats) {
        dim3 g((M + 63) / 64, (Nout + 15) / 16);
        gemm_wmma_kernel<<<g, 128, 0, stream>>>(Xp, M, Kd, W, bias, Y, Nout, stats);
    };

    auto smlp = [&](const float* Xp, int M, int Kd, int Nout, int pb,
                    float* Y, int mode, const float* resid) {
        zero_kernel<<<1, 256, 0, stream>>>(STB, 256);
        gemm(Xp, M, Kd, P(pb), P(pb + 1), Y, Nout, STB);
        finalize_bn_kernel<<<1, 128, 0, stream>>>(STB, P(pb + 3), P(pb + 2),
                                                  Nout, 1.0f / (float)M);
        int tot = M * Nout;
        bn_act_kernel<<<(tot + 255) / 256, 256, 0, stream>>>(Y, STB, tot,
                                                             Nout - 1, mode, resid);
    };

    auto lfa = [&](int pb, const float* xp, const float* posp, int bnShift,
                   int Mtot, int d, float* out) {
        int d2 = d / 2;
        zero_kernel<<<1, 256, 0, stream>>>(STA, 256);
        lfa_enc_stats_kernel<<<(Mtot * 16 + 255) / 256, 256, 0, stream>>>(
            posp, IDX, bnShift, Mtot, P(pb + 1), P(pb + 2), d2, STA);
        finalize_bn_kernel<<<1, 128, 0, stream>>>(STA, P(pb + 4), P(pb + 3), d2,
                                                  1.0f / (float)(Mtot * 16));
        zero_kernel<<<1, 256, 0, stream>>>(STB, 256);
        lfa_main_kernel<<<(Mtot + 3) / 4, 128, 0, stream>>>(
            xp, posp, IDX, bnShift, Mtot, d, P(pb + 1), P(pb + 2), STA,
            P(pb), P(pb + 5), P(pb + 6), out, STB);
        finalize_bn_kernel<<<1, 128, 0, stream>>>(STB, P(pb + 8), P(pb + 7), d,
                                                  1.0f / (float)Mtot);
        int tot = Mtot * d;
        bn_act_kernel<<<(tot + 255) / 256, 256, 0, stream>>>(out, STB, tot,
                                                             d - 1, 1, nullptr);
    };

    auto blockf = [&](int pb, const float* Xp, const float* posp, int Bn,
                      int bnShift, int di, int dof, float* out) {
        int Mtot = 16 * Bn;
        knn_kernel<<<dim3((Bn + 127) / 128, 16), 128, 0, stream>>>(posp, Bn, IDX);
        smlp(Xp, Mtot, di, dof, pb + 26, SC, 0, nullptr);          // shortcut
        smlp(Xp, Mtot, di, dof / 8, pb + 18, TMPA, 1, nullptr);    // mlp1
        lfa(pb + 0, TMPA, posp, bnShift, Mtot, dof / 4, TMPB);     // lfa1
        lfa(pb + 9, TMPB, posp, bnShift, Mtot, dof / 2, TMPA);     // lfa2
        smlp(TMPA, Mtot, dof / 2, dof, pb + 22, out, 2, SC);       // mlp2+resid
    };

    // fc0: plain linear 6->8 (no BN)
    gemm(X0, 32768, 6, P(66), P(67), HA, 8, nullptr);
    // block1: N=2048, 8->32
    blockf(2, HA, POS, 2048, 11, 8, 32, HB);
    // decimate to N=512
    slice_rows_kernel<<<(16 * 512 * 32 + 255) / 256, 256, 0, stream>>>(HB, 2048, HA, 512, 32);
    slice_rows_kernel<<<(16 * 512 * 3 + 255) / 256, 256, 0, stream>>>(POS, 2048, POS2, 512, 3);
    // block2: N=512, 32->128
    blockf(32, HA, POS2, 512, 9, 32, 128, HB);
    // decimate to N=128
    slice_rows_kernel<<<(16 * 128 * 128 + 255) / 256, 256, 0, stream>>>(HB, 512, HA, 128, 128);
    // mlp1: 128->128 over 2048 rows
    smlp(HA, 2048, 128, 128, 70, TMPA, 1, nullptr);
    // global max pool
    maxpool_kernel<<<16, 128, 0, stream>>>(TMPA, 128, 128, G1);
    // end0: 128->32 over 16 rows
    smlp(G1, 16, 128, 32, 62, G2, 1, nullptr);
    // fc_end: 32->40 (no BN)
    gemm(G2, 16, 32, P(68), P(69), G3, 40, nullptr);
    // log softmax
    logsoftmax_kernel<<<1, 32, 0, stream>>>(G3, (float*)d_out);
}